// KCompetitive_22703197127201
// MI455X (gfx1250) — compile-verified
//
#include <hip/hip_runtime.h>

// KCompetitive (k_comp_tanh training branch), MI455X / gfx1250.
// Bandwidth-bound: 256 MiB total traffic -> ~11.5 us at 23.3 TB/s.
// One wave32x8 workgroup per row; row staged to LDS via CDNA5 async
// global->LDS b128 copies (also performs the coalesced->chunked layout
// shuffle); chunk then cached in registers as monotonic uint keys; top-k
// via LDS radix select (4x 8-bit passes, single-wave shfl scans).

#define ROW_C   2048
#define NTH     256
#define EPT     8          // elements per thread (processing layout, chunked)
#define K_P     32
#define K_N     32
#define KFACTOR 6.26f

typedef __attribute__((ext_vector_type(4))) float f4;
typedef __attribute__((ext_vector_type(4))) int   i4;
typedef __attribute__((address_space(1))) i4 as1_i4;   // global payload ptr
typedef __attribute__((address_space(3))) i4 as3_i4;   // LDS payload ptr

#if defined(__gfx1250__) && \
    __has_builtin(__builtin_amdgcn_global_load_async_to_lds_b128) && \
    __has_builtin(__builtin_amdgcn_s_wait_asynccnt)
#define KC_ASYNC 1
#else
#define KC_ASYNC 0
#endif

__device__ __forceinline__ unsigned keyOf(float v, bool neg) {
    // monotonic uint key of the relevant magnitude; 0 for non-candidates
    float m = neg ? -v : v;
    return (m > 0.0f) ? __float_as_uint(m) : 0u;
}

// Deterministic block-wide float sum: shfl tree per wave, fixed-order combine.
__device__ __forceinline__ float blockSum(float v, float* s_w, int t) {
#pragma unroll
    for (int m = 16; m > 0; m >>= 1) v += __shfl_xor(v, m, 32);
    __syncthreads();                       // protect previous s_w readers
    if ((t & 31) == 0) s_w[t >> 5] = v;
    __syncthreads();
    float r = 0.0f;
#pragma unroll
    for (int w = 0; w < 8; ++w) r += s_w[w];  // fixed order -> deterministic
    __syncthreads();
    return r;
}

// In-place inclusive SUFFIX sum of s_hist[0..255], done by wave 0 only
// (8 bins per lane + shfl_down cross-lane scan; no internal barriers).
__device__ __forceinline__ void waveSuffixScan256(unsigned* s_hist, int t) {
    if (t < 32) {
        unsigned v[8];
#pragma unroll
        for (int i = 0; i < 8; ++i) v[i] = s_hist[t * 8 + i];
#pragma unroll
        for (int i = 6; i >= 0; --i) v[i] += v[i + 1];   // local suffix
        unsigned tot = v[0], run = tot;
#pragma unroll
        for (int d = 1; d < 32; d <<= 1) {
            unsigned o = __shfl_down(run, d, 32);
            if (t + d < 32) run += o;                    // inclusive suffix of totals
        }
        unsigned excl = run - tot;                       // from higher lanes only
#pragma unroll
        for (int i = 0; i < 8; ++i) s_hist[t * 8 + i] = v[i] + excl;
    }
}

// In-place inclusive PREFIX sum of s_hist[0..255], wave 0 only.
__device__ __forceinline__ void wavePrefixScan256(unsigned* s_hist, int t) {
    if (t < 32) {
        unsigned v[8];
#pragma unroll
        for (int i = 0; i < 8; ++i) v[i] = s_hist[t * 8 + i];
#pragma unroll
        for (int i = 1; i < 8; ++i) v[i] += v[i - 1];    // local prefix
        unsigned tot = v[7], run = tot;
#pragma unroll
        for (int d = 1; d < 32; d <<= 1) {
            unsigned o = __shfl_up(run, d, 32);
            if (t >= d) run += o;                        // inclusive prefix of totals
        }
        unsigned excl = run - tot;                       // from lower lanes only
#pragma unroll
        for (int i = 0; i < 8; ++i) s_hist[t * 8 + i] = v[i] + excl;
    }
}

// Radix select over register-resident keys: threshold key T of the kwant-th
// largest key, and r = number of T-valued elements still needed (ties taken
// in lowest-index order).
__device__ __forceinline__ void radixTop(const unsigned* k, unsigned* s_hist,
                                         unsigned* s_bc, int t,
                                         unsigned kwant, unsigned& T, unsigned& r) {
    unsigned prefix = 0u, remaining = kwant;
#pragma unroll
    for (int shift = 24; shift >= 0; shift -= 8) {
        const unsigned himask = (shift == 24) ? 0u : (0xFFFFFFFFu << (shift + 8));
        s_hist[t] = 0u;
        if (t < 2) s_bc[t] = 0u;
        __syncthreads();
#pragma unroll
        for (int e = 0; e < EPT; ++e) {
            if ((k[e] & himask) == (prefix & himask))
                atomicAdd(&s_hist[(k[e] >> shift) & 255u], 1u);
        }
        __syncthreads();
        waveSuffixScan256(s_hist, t);
        __syncthreads();
        unsigned suf  = s_hist[t];
        unsigned sufN = (t < 255) ? s_hist[t + 1] : 0u;
        if (suf >= remaining && sufN < remaining) {
            s_bc[0] = (unsigned)t;   // winning digit
            s_bc[1] = sufN;          // count with strictly greater digit
        }
        __syncthreads();
        prefix    |= s_bc[0] << shift;
        remaining -= s_bc[1];
        __syncthreads();
    }
    T = prefix;
    r = remaining;
}

// Stable rank: number of elements with key==T at indices before this thread's chunk.
__device__ __forceinline__ unsigned tieBefore(const unsigned* k, unsigned* s_hist,
                                              int t, unsigned T) {
    unsigned myEq = 0;
#pragma unroll
    for (int e = 0; e < EPT; ++e) myEq += (k[e] == T) ? 1u : 0u;
    __syncthreads();                 // protect previous s_hist readers
    s_hist[t] = myEq;
    __syncthreads();
    wavePrefixScan256(s_hist, t);
    __syncthreads();
    unsigned before = s_hist[t] - myEq;   // exclusive prefix for my chunk
    __syncthreads();
    return before;
}

__global__ __launch_bounds__(NTH) void kcomp_kernel(const float* __restrict__ x,
                                                    float* __restrict__ out) {
    __shared__ float    s_row[ROW_C];
    __shared__ unsigned s_hist[256];
    __shared__ unsigned s_bc[2];
    __shared__ float    s_w[8];

    const int t = threadIdx.x;
    const long long row = blockIdx.x;
    const float* __restrict__ gx = x + row * (long long)ROW_C;
    float* __restrict__ gy = out + row * (long long)ROW_C;

    // ---- stage row into LDS (coalesced; CDNA5 async-to-LDS when available) ----
#if KC_ASYNC
    {
        // global side: generic and AS1 share address values -> inttoptr is exact
        as1_i4* g0 = (as1_i4*)(unsigned long long)(gx + t * 4);
        as1_i4* g1 = (as1_i4*)(unsigned long long)(gx + 1024 + t * 4);
        // LDS side: must be a true generic->AS3 addrspacecast (32-bit offset)
        as3_i4* l0 = (as3_i4*)(&s_row[t * 4]);
        as3_i4* l1 = (as3_i4*)(&s_row[1024 + t * 4]);
        __builtin_amdgcn_global_load_async_to_lds_b128(g0, l0, 0, 0);
        __builtin_amdgcn_global_load_async_to_lds_b128(g1, l1, 0, 0);
        __builtin_amdgcn_s_wait_asynccnt(0);
    }
#else
    {
        f4 a = *(const f4*)(gx + t * 4);
        f4 b = *(const f4*)(gx + 1024 + t * 4);
        *(f4*)(&s_row[t * 4])        = a;
        *(f4*)(&s_row[1024 + t * 4]) = b;
    }
#endif
    __syncthreads();

    // ---- read my chunk ONCE, keep monotonic keys in registers ----
    f4 c0 = *(const f4*)(&s_row[t * EPT]);
    f4 c1 = *(const f4*)(&s_row[t * EPT + 4]);
    float v[EPT] = {c0.x, c0.y, c0.z, c0.w, c1.x, c1.y, c1.z, c1.w};
    unsigned kp[EPT], kn[EPT];
#pragma unroll
    for (int e = 0; e < EPT; ++e) {
        kp[e] = keyOf(v[e], false);   // bits of relu(x)
        kn[e] = keyOf(v[e], true);    // bits of relu(-x)
    }

    // ---- total positive / negative energies (uint_as_float(0) == 0.0) ----
    float lp = 0.0f, ln = 0.0f;
#pragma unroll
    for (int e = 0; e < EPT; ++e) {
        lp += __uint_as_float(kp[e]);
        ln += __uint_as_float(kn[e]);
    }
    const float sumP = blockSum(lp, s_w, t);
    const float sumN = blockSum(ln, s_w, t);

    // ---- top-k thresholds via LDS radix select ----
    unsigned Tp, rpTake, Tn, rnTake;
    radixTop(kp, s_hist, s_bc, t, K_P, Tp, rpTake);
    radixTop(kn, s_hist, s_bc, t, K_N, Tn, rnTake);

    // ---- winner energies ----
    float lwp = 0.0f, lwn = 0.0f;
#pragma unroll
    for (int e = 0; e < EPT; ++e) {
        if (kp[e] > Tp) lwp += __uint_as_float(kp[e]);
        if (kn[e] > Tn) lwn += __uint_as_float(kn[e]);
    }
    const float sumWp = blockSum(lwp, s_w, t) + (float)rpTake * __uint_as_float(Tp);
    const float sumWn = blockSum(lwn, s_w, t) + (float)rnTake * __uint_as_float(Tn);

    const float Ptmp = KFACTOR * (sumP - sumWp);   // amplified loser energy (pos)
    const float Ntmp = KFACTOR * (sumN - sumWn);   // amplified loser energy (neg)

    // ---- stable tie ranks (lowest index wins, matching lax.top_k) ----
    unsigned cp = tieBefore(kp, s_hist, t, Tp);
    unsigned cn = tieBefore(kn, s_hist, t, Tn);

    // ---- emit P_reset - N_reset ----
    float o[EPT];
#pragma unroll
    for (int e = 0; e < EPT; ++e) {
        float r = 0.0f;
        if (kp[e] > Tp) {
            r = __uint_as_float(kp[e]) + Ptmp;
        } else if (kp[e] == Tp) {
            if (cp < rpTake) r = __uint_as_float(kp[e]) + Ptmp;
            ++cp;
        }
        if (kn[e] > Tn) {
            r = -(__uint_as_float(kn[e]) + Ntmp);
        } else if (kn[e] == Tn) {
            if (cn < rnTake) r = -(__uint_as_float(kn[e]) + Ntmp);
            ++cn;
        }
        o[e] = r;
    }
    f4 o0 = {o[0], o[1], o[2], o[3]};
    f4 o1 = {o[4], o[5], o[6], o[7]};
    __builtin_nontemporal_store(o0, (f4*)(gy + t * EPT));
    __builtin_nontemporal_store(o1, (f4*)(gy + t * EPT + 4));
}

extern "C" void kernel_launch(void* const* d_in, const int* in_sizes, int n_in,
                              void* d_out, int out_size, void* d_ws, size_t ws_size,
                              hipStream_t stream) {
    (void)n_in; (void)out_size; (void)d_ws; (void)ws_size;
    const float* x = (const float*)d_in[0];
    float* out = (float*)d_out;
    const int rows = in_sizes[0] / ROW_C;   // 16384 x 2048 fp32
    if (rows <= 0) return;
    kcomp_kernel<<<dim3((unsigned)rows), dim3(NTH), 0, stream>>>(x, out);
}